// CausalGCN_55155970015565
// MI455X (gfx1250) — compile-verified
//
#include <hip/hip_runtime.h>
#include <math.h>

// ---------------------------------------------------------------------------
// CausalGCN forward for MI455X (gfx1250, wave32, WMMA).
// Dense GEMMs (~5.7 GFLOP fp32) run on V_WMMA_F32_16X16X4_F32 with BN folded
// into the weights; the bandwidth-dominant part is the 5 edge propagations
// (~2.6 GB -> ~110us @ 23.3 TB/s), whose f32 atomics land in the 192MB L2.
// ---------------------------------------------------------------------------

#define N_NODES  100000
#define N_EDGES  1000000
#define N_FEAT   128
#define HIDDEN   64
#define N_GRAPHS 512
#define N_CLASSES 10
#define BN_EPS  1e-5f
#define BN_BETA 1e-4f

typedef __attribute__((ext_vector_type(2))) float v2f;
typedef __attribute__((ext_vector_type(8))) float v8f;

// ---- per-column sum / sum-of-squares (blockDim.x == F, coalesced rows) ----
__global__ void colstats_kernel(const float* __restrict__ A, int N, int F,
                                float* __restrict__ sums, float* __restrict__ sumsq) {
  const int c = threadIdx.x;
  float s = 0.f, q = 0.f;
  for (int r = blockIdx.x; r < N; r += gridDim.x) {
    float v = A[(size_t)r * F + c];
    s += v; q += v * v;
  }
  atomicAdd(&sums[c], s);
  atomicAdd(&sumsq[c], q);
}

// ---- fold BN (gamma=1, beta=BN_BETA) + optional bias into weight ----------
// bn(x)@W + eb  ==  x @ (diag(scale)W) + vout
__global__ void fold_bn_kernel(const float* __restrict__ sums,
                               const float* __restrict__ sumsq, float invN,
                               int K, int NN, const float* __restrict__ W,
                               const float* __restrict__ extra_b,
                               float* __restrict__ Wout, float* __restrict__ vout) {
  __shared__ float s_ms[N_FEAT];  // mean*scale
  __shared__ float s_s[N_FEAT];   // scale
  const int j = threadIdx.x;      // blockDim.x == NN
  for (int k = j; k < K; k += blockDim.x) {
    float mean = sums[k] * invN;
    float var  = sumsq[k] * invN - mean * mean;
    float sc   = rsqrtf(var + BN_EPS);
    s_ms[k] = mean * sc;
    s_s[k]  = sc;
  }
  __syncthreads();
  float acc = extra_b ? extra_b[j] : 0.f;
  for (int k = 0; k < K; ++k) {
    float w = W[(size_t)k * NN + j];
    Wout[(size_t)k * NN + j] = s_s[k] * w;
    acc += (BN_BETA - s_ms[k]) * w;
  }
  vout[j] = acc;
}

// ---- WMMA f32 GEMM: C[M x 64] = A[M x K] @ Wt[K x 64] + v, optional ReLU --
// One wave per 16-row tile; the A fragment is reused across the four 16-col
// tiles. Per ISA layouts (16x16x4 f32): lane L holds A[m=L%16][k0..k0+1] and
// B[k0..k0+1][n=L%16] with k0 = (L/16)*2 + kk; C/D elem e at lane L is
// row (e + 8*(L/16)), col (L%16).
__global__ __launch_bounds__(128) void gemm_wmma_kernel(
    const float* __restrict__ A, const float* __restrict__ Wt,
    const float* __restrict__ v, float* __restrict__ C,
    int M, int K, int doRelu) {
  const int lane = threadIdx.x & 31;
  const int wave = threadIdx.x >> 5;
  const int rowTile = blockIdx.x * 4 + wave;
  if (rowTile >= (M >> 4)) return;          // wave-uniform: EXEC stays full
  const int mrow = (rowTile << 4) + (lane & 15);
  const int kOff = (lane >> 4) << 1;        // 0 or 2
  const int n0   = lane & 15;
  v8f acc[4] = {};
  for (int kk = 0; kk < K; kk += 4) {
    const int k0 = kk + kOff;
    v2f a = *(const v2f*)(A + (size_t)mrow * K + k0);   // 8B aligned (k0 even)
    const float* w0 = Wt + (size_t)k0 * HIDDEN + n0;
#pragma unroll
    for (int t = 0; t < 4; ++t) {
      v2f b;
      b.x = w0[t * 16];
      b.y = w0[HIDDEN + t * 16];
      acc[t] = __builtin_amdgcn_wmma_f32_16x16x4_f32(
          false, a, false, b, (short)0, acc[t], false, false);
    }
  }
  const int rowBase = (rowTile << 4) + ((lane >> 4) << 3);
#pragma unroll
  for (int t = 0; t < 4; ++t) {
    const int n = t * 16 + n0;
    const float bias = v[n];
#pragma unroll
    for (int e = 0; e < 8; ++e) {
      float val = acc[t][e] + bias;
      if (doRelu) val = fmaxf(val, 0.f);
      C[(size_t)(rowBase + e) * HIDDEN + n] = val;
    }
  }
}

// ---- degree / inverse-sqrt --------------------------------------------------
__global__ void deg_init_kernel(float* deg, int N) {
  int i = blockIdx.x * blockDim.x + threadIdx.x;
  if (i < N) deg[i] = 1.0f;                 // self-loop weight
}
__global__ void deg_accum_kernel(const int* __restrict__ row,
                                 const float* __restrict__ att, int comp,
                                 int E, float* __restrict__ deg) {
  int e = blockIdx.x * blockDim.x + threadIdx.x;
  if (e >= E) return;
  float w = att ? att[(size_t)e * 2 + comp] : 1.0f;
  atomicAdd(&deg[row[e]], w);
}
__global__ void make_dinv_kernel(const float* __restrict__ deg,
                                 float* __restrict__ dinv, int N) {
  int i = blockIdx.x * blockDim.x + threadIdx.x;
  if (i < N) dinv[i] = deg[i] > 0.f ? rsqrtf(deg[i]) : 0.f;
}

// ---- edge scatter: acc[col] += dinv[row]*w*dinv[col] * m[row] ---------------
__global__ void scatter_kernel(const float* __restrict__ m,
                               const int* __restrict__ row,
                               const int* __restrict__ col,
                               const float* __restrict__ att, int comp,
                               const float* __restrict__ dinv,
                               float* __restrict__ acc, int E) {
  int idx = blockIdx.x * blockDim.x + threadIdx.x;
  int e = idx >> 6, j = idx & 63;
  if (e >= E) return;
  int r = row[e], c = col[e];
  float w = att ? att[(size_t)e * 2 + comp] : 1.0f;
  float coef = dinv[r] * w * dinv[c];
  atomicAdd(&acc[(size_t)c * HIDDEN + j], coef * m[(size_t)r * HIDDEN + j]);
}

// ---- self-loop + bias + ReLU ------------------------------------------------
__global__ void finish_gcn_kernel(const float* __restrict__ acc,
                                  const float* __restrict__ m,
                                  const float* __restrict__ dinv,
                                  const float* __restrict__ b,
                                  float* __restrict__ out, int N) {
  int idx = blockIdx.x * blockDim.x + threadIdx.x;
  int i = idx >> 6, j = idx & 63;
  if (i >= N) return;
  float di = dinv[i];
  float val = acc[idx] + di * di * m[idx] + b[j];
  out[idx] = fmaxf(val, 0.f);
}

// ---- edge attention: softmax([h[row],h[col]] @ W_ea + b_ea) over 2 ----------
__global__ void edge_att_kernel(const float* __restrict__ h,
                                const int* __restrict__ row,
                                const int* __restrict__ col,
                                const float* __restrict__ Wea,
                                const float* __restrict__ bea,
                                float* __restrict__ att, int E) {
  int e = blockIdx.x * blockDim.x + threadIdx.x;
  if (e >= E) return;
  const float* hr = h + (size_t)row[e] * HIDDEN;
  const float* hc = h + (size_t)col[e] * HIDDEN;
  float l0 = bea[0], l1 = bea[1];
  for (int j = 0; j < HIDDEN; ++j) {
    float a = hr[j], c = hc[j];
    l0 += a * Wea[j * 2 + 0] + c * Wea[(HIDDEN + j) * 2 + 0];
    l1 += a * Wea[j * 2 + 1] + c * Wea[(HIDDEN + j) * 2 + 1];
  }
  float mx = fmaxf(l0, l1);
  float e0 = __expf(l0 - mx), e1 = __expf(l1 - mx);
  float inv = 1.f / (e0 + e1);
  att[(size_t)e * 2 + 0] = e0 * inv;
  att[(size_t)e * 2 + 1] = e1 * inv;
}

// ---- node attention + split into xc / xo ------------------------------------
__global__ void node_att_split_kernel(const float* __restrict__ h,
                                      const float* __restrict__ Wna,
                                      const float* __restrict__ bna,
                                      float* __restrict__ xc,
                                      float* __restrict__ xo, int N) {
  int i = blockIdx.x * blockDim.x + threadIdx.x;
  if (i >= N) return;
  const float* hi = h + (size_t)i * HIDDEN;
  float l0 = bna[0], l1 = bna[1];
  for (int j = 0; j < HIDDEN; ++j) {
    float a = hi[j];
    l0 += a * Wna[j * 2 + 0];
    l1 += a * Wna[j * 2 + 1];
  }
  float mx = fmaxf(l0, l1);
  float e0 = __expf(l0 - mx), e1 = __expf(l1 - mx);
  float inv = 1.f / (e0 + e1);
  float p0 = e0 * inv, p1 = e1 * inv;
  for (int j = 0; j < HIDDEN; ++j) {
    float a = hi[j];
    xc[(size_t)i * HIDDEN + j] = p0 * a;
    xo[(size_t)i * HIDDEN + j] = p1 * a;
  }
}

// ---- global add pool --------------------------------------------------------
__global__ void pool_kernel(const float* __restrict__ x,
                            const int* __restrict__ batch,
                            float* __restrict__ g, int N) {
  int idx = blockIdx.x * blockDim.x + threadIdx.x;
  int i = idx >> 6, j = idx & 63;
  if (i >= N) return;
  atomicAdd(&g[(size_t)batch[i] * HIDDEN + j], x[idx]);
}

__global__ void add_kernel(const float* a, const float* b, float* c, int n) {
  int i = blockIdx.x * blockDim.x + threadIdx.x;
  if (i < n) c[i] = a[i] + b[i];
}

// ---- final 64 -> 10 logits (BN already folded) + log_softmax ----------------
__global__ void logits_kernel(const float* __restrict__ h,
                              const float* __restrict__ Wt,
                              const float* __restrict__ v,
                              float* __restrict__ out, int G) {
  int g = blockIdx.x * blockDim.x + threadIdx.x;
  if (g >= G) return;
  float l[N_CLASSES];
  for (int c = 0; c < N_CLASSES; ++c) l[c] = v[c];
  const float* hg = h + (size_t)g * HIDDEN;
  for (int k = 0; k < HIDDEN; ++k) {
    float x = hg[k];
    for (int c = 0; c < N_CLASSES; ++c) l[c] += x * Wt[k * N_CLASSES + c];
  }
  float mx = l[0];
  for (int c = 1; c < N_CLASSES; ++c) mx = fmaxf(mx, l[c]);
  float s = 0.f;
  for (int c = 0; c < N_CLASSES; ++c) s += __expf(l[c] - mx);
  float lse = __logf(s) + mx;
  for (int c = 0; c < N_CLASSES; ++c) out[(size_t)g * N_CLASSES + c] = l[c] - lse;
}

// ---------------------------------------------------------------------------
extern "C" void kernel_launch(void* const* d_in, const int* in_sizes, int n_in,
                              void* d_out, int out_size, void* d_ws, size_t ws_size,
                              hipStream_t stream) {
  (void)in_sizes; (void)n_in; (void)out_size; (void)ws_size;
  const float* x       = (const float*)d_in[0];
  const int*   ei      = (const int*)d_in[1];
  const int*   row     = ei;
  const int*   col     = ei + N_EDGES;
  const int*   batch   = (const int*)d_in[2];
  const float* W_feat  = (const float*)d_in[3];
  const float* W_convs = (const float*)d_in[4];
  const float* b_convs = (const float*)d_in[5];
  const float* W_ea    = (const float*)d_in[6];
  const float* b_ea    = (const float*)d_in[7];
  const float* W_na    = (const float*)d_in[8];
  const float* b_na    = (const float*)d_in[9];
  const float* W_ctx   = (const float*)d_in[10];
  const float* b_ctx   = (const float*)d_in[11];
  const float* W_obj   = (const float*)d_in[12];
  const float* b_obj   = (const float*)d_in[13];
  const float* W1h[3]  = {(const float*)d_in[14], (const float*)d_in[18], (const float*)d_in[22]};
  const float* b1h[3]  = {(const float*)d_in[15], (const float*)d_in[19], (const float*)d_in[23]};
  const float* W2h[3]  = {(const float*)d_in[16], (const float*)d_in[20], (const float*)d_in[24]};
  const float* b2h[3]  = {(const float*)d_in[17], (const float*)d_in[21], (const float*)d_in[25]};

  // ---- workspace carve ----
  size_t off = 0;
  auto carve = [&](size_t elems) -> float* {
    float* p = (float*)((char*)d_ws + off);
    off += ((elems * sizeof(float)) + 255) & ~(size_t)255;
    return p;
  };
  float* hb    = carve((size_t)N_NODES * HIDDEN);   // node features
  float* mbuf  = carve((size_t)N_NODES * HIDDEN);   // transformed features
  float* accb  = carve((size_t)N_NODES * HIDDEN);   // scatter accumulator
  float* xc    = carve((size_t)N_NODES * HIDDEN);
  float* xo    = carve((size_t)N_NODES * HIDDEN);
  float* att   = carve((size_t)N_EDGES * 2);
  float* degu  = carve(N_NODES);
  float* dinvu = carve(N_NODES);
  float* dega  = carve(N_NODES);
  float* dinva = carve(N_NODES);
  float* gc    = carve((size_t)N_GRAPHS * HIDDEN);
  float* go    = carve((size_t)N_GRAPHS * HIDDEN);
  float* gs    = carve((size_t)N_GRAPHS * HIDDEN);
  float* r1    = carve((size_t)N_GRAPHS * HIDDEN);
  float* sums  = carve(N_FEAT);
  float* sq    = carve(N_FEAT);
  float* Wf    = carve((size_t)N_FEAT * HIDDEN);    // folded weight
  float* vf    = carve(HIDDEN);                     // folded bias

  auto nblk = [](long n, int b) { return (int)((n + b - 1) / b); };

  auto stats = [&](const float* A, int N, int F) {
    hipMemsetAsync(sums, 0, (size_t)F * sizeof(float), stream);
    hipMemsetAsync(sq,   0, (size_t)F * sizeof(float), stream);
    colstats_kernel<<<512, F, 0, stream>>>(A, N, F, sums, sq);
  };
  auto gemm = [&](const float* A, float* C, int M, int K, int relu) {
    gemm_wmma_kernel<<<nblk(M / 16, 4), 128, 0, stream>>>(A, Wf, vf, C, M, K, relu);
  };

  // ---- Stage 1: h = relu(bn(x) @ W_feat) ----
  stats(x, N_NODES, N_FEAT);
  fold_bn_kernel<<<1, HIDDEN, 0, stream>>>(sums, sq, 1.f / N_NODES, N_FEAT, HIDDEN,
                                           W_feat, nullptr, Wf, vf);
  gemm(x, hb, N_NODES, N_FEAT, 1);

  // ---- unit-weight degrees (shared by the 3 conv layers) ----
  deg_init_kernel<<<nblk(N_NODES, 256), 256, 0, stream>>>(degu, N_NODES);
  deg_accum_kernel<<<nblk(N_EDGES, 256), 256, 0, stream>>>(row, nullptr, 0, N_EDGES, degu);
  make_dinv_kernel<<<nblk(N_NODES, 256), 256, 0, stream>>>(degu, dinvu, N_NODES);

  // ---- 3 GCN layers: h = relu(gcn(bn(h))) ----
  for (int i = 0; i < 3; ++i) {
    stats(hb, N_NODES, HIDDEN);
    fold_bn_kernel<<<1, HIDDEN, 0, stream>>>(sums, sq, 1.f / N_NODES, HIDDEN, HIDDEN,
                                             W_convs + (size_t)i * HIDDEN * HIDDEN,
                                             nullptr, Wf, vf);
    gemm(hb, mbuf, N_NODES, HIDDEN, 0);
    hipMemsetAsync(accb, 0, (size_t)N_NODES * HIDDEN * sizeof(float), stream);
    scatter_kernel<<<nblk((long)N_EDGES * 64, 256), 256, 0, stream>>>(
        mbuf, row, col, nullptr, 0, dinvu, accb, N_EDGES);
    finish_gcn_kernel<<<nblk((long)N_NODES * 64, 256), 256, 0, stream>>>(
        accb, mbuf, dinvu, b_convs + (size_t)i * HIDDEN, hb, N_NODES);
  }

  // ---- attention ----
  edge_att_kernel<<<nblk(N_EDGES, 256), 256, 0, stream>>>(hb, row, col, W_ea, b_ea,
                                                          att, N_EDGES);
  node_att_split_kernel<<<nblk(N_NODES, 256), 256, 0, stream>>>(hb, W_na, b_na,
                                                                xc, xo, N_NODES);

  // ---- attention-weighted GCN branch + pool ----
  auto run_branch = [&](float* xin, int comp, const float* W, const float* b,
                        float* gout) {
    stats(xin, N_NODES, HIDDEN);
    fold_bn_kernel<<<1, HIDDEN, 0, stream>>>(sums, sq, 1.f / N_NODES, HIDDEN, HIDDEN,
                                             W, nullptr, Wf, vf);
    gemm(xin, mbuf, N_NODES, HIDDEN, 0);
    deg_init_kernel<<<nblk(N_NODES, 256), 256, 0, stream>>>(dega, N_NODES);
    deg_accum_kernel<<<nblk(N_EDGES, 256), 256, 0, stream>>>(row, att, comp, N_EDGES, dega);
    make_dinv_kernel<<<nblk(N_NODES, 256), 256, 0, stream>>>(dega, dinva, N_NODES);
    hipMemsetAsync(accb, 0, (size_t)N_NODES * HIDDEN * sizeof(float), stream);
    scatter_kernel<<<nblk((long)N_EDGES * 64, 256), 256, 0, stream>>>(
        mbuf, row, col, att, comp, dinva, accb, N_EDGES);
    finish_gcn_kernel<<<nblk((long)N_NODES * 64, 256), 256, 0, stream>>>(
        accb, mbuf, dinva, b, xin, N_NODES);
    hipMemsetAsync(gout, 0, (size_t)N_GRAPHS * HIDDEN * sizeof(float), stream);
    pool_kernel<<<nblk((long)N_NODES * 64, 256), 256, 0, stream>>>(xin, batch, gout,
                                                                   N_NODES);
  };
  run_branch(xc, 0, W_ctx, b_ctx, gc);
  run_branch(xo, 1, W_obj, b_obj, go);

  add_kernel<<<nblk(N_GRAPHS * HIDDEN, 256), 256, 0, stream>>>(gc, go, gs,
                                                               N_GRAPHS * HIDDEN);

  // ---- readout heads ----
  float* out = (float*)d_out;
  auto run_head = [&](const float* gin, int h, float* outp) {
    stats(gin, N_GRAPHS, HIDDEN);
    fold_bn_kernel<<<1, HIDDEN, 0, stream>>>(sums, sq, 1.f / N_GRAPHS, HIDDEN, HIDDEN,
                                             W1h[h], b1h[h], Wf, vf);
    gemm(gin, r1, N_GRAPHS, HIDDEN, 1);
    stats(r1, N_GRAPHS, HIDDEN);
    fold_bn_kernel<<<1, N_CLASSES, 0, stream>>>(sums, sq, 1.f / N_GRAPHS, HIDDEN,
                                                N_CLASSES, W2h[h], b2h[h], Wf, vf);
    logits_kernel<<<nblk(N_GRAPHS, 256), 256, 0, stream>>>(r1, Wf, vf, outp, N_GRAPHS);
  };
  run_head(gc, 0, out);
  run_head(go, 1, out + N_GRAPHS * N_CLASSES);
  run_head(gs, 2, out + 2 * N_GRAPHS * N_CLASSES);
}